// Net_4913442587175
// MI455X (gfx1250) — compile-verified
//
#include <hip/hip_runtime.h>
#include <hip/hip_bf16.h>

typedef __attribute__((ext_vector_type(2))) float v2f;
typedef __attribute__((ext_vector_type(8))) float v8f;

// ---------------------------------------------------------------------------
// fills
// ---------------------------------------------------------------------------
__global__ void k_fill(float* __restrict__ p, float v, int n) {
    int i = blockIdx.x * blockDim.x + threadIdx.x;
    if (i < n) p[i] = v;
}

__global__ void k_fill4(float4* __restrict__ p, float v, int n4) {
    int i = blockIdx.x * blockDim.x + threadIdx.x;
    if (i < n4) p[i] = make_float4(v, v, v, v);
}

// ---------------------------------------------------------------------------
// degree / norm helpers
// ---------------------------------------------------------------------------
__global__ void k_count_deg(const int* __restrict__ dst, float* __restrict__ deg, int E) {
    int e = blockIdx.x * blockDim.x + threadIdx.x;
    if (e < E) atomicAdd(&deg[dst[e]], 1.0f);
}

__global__ void k_rsqrt_inplace(float* __restrict__ p, int n) {
    int i = blockIdx.x * blockDim.x + threadIdx.x;
    if (i < n) p[i] = rsqrtf(fmaxf(p[i], 1.0f));
}

// ---------------------------------------------------------------------------
// fp32 WMMA GEMM:  C[M x Nc] = act(A[M x K] @ B[K x Nc] + bias)
// One wave per 16x16 output tile via V_WMMA_F32_16X16X4_F32.
// Hot loop is branch-free: A loads row-clamped (always in bounds), B loads
// column-clamped with a 0/1 float mask instead of EXEC-diverging guards.
// K remainder (K%4) handled once in a guarded tail.
// ---------------------------------------------------------------------------
__global__ void k_gemm_wmma_f32(const float* __restrict__ A,
                                const float* __restrict__ B,
                                const float* __restrict__ bias,
                                float* __restrict__ C,
                                int M, int K, int Nc, int relu) {
    const int lane = threadIdx.x & 31;
    const int wave = (blockIdx.x * blockDim.x + threadIdx.x) >> 5;
    const int ntile = (Nc + 15) >> 4;
    const int mtile = (M + 15) >> 4;
    if (wave >= ntile * mtile) return;              // wave-uniform exit

    const int tm   = wave / ntile;
    const int tn   = wave - tm * ntile;
    const int row0 = tm << 4;
    const int col0 = tn << 4;

    const int   mlo   = lane & 15;
    const int   half  = lane >> 4;                  // 0: K pair {0,1}, 1: {2,3}
    const int   rowA  = min(row0 + mlo, M - 1);
    const int   n     = col0 + mlo;
    const bool  nok   = (n < Nc);
    const int   ncl   = nok ? n : 0;
    const float nmask = nok ? 1.0f : 0.0f;

    const float* __restrict__ Arow = A + (size_t)rowA * K;
    const int K4 = K & ~3;

    v8f acc = {};
#pragma unroll 2
    for (int k0 = 0; k0 < K4; k0 += 4) {
        const int ka = k0 + (half << 1);
        v2f a, b;
        a.x = Arow[ka];                              // adjacent -> b64 load
        a.y = Arow[ka + 1];
        b.x = B[(size_t)ka * Nc + ncl] * nmask;
        b.y = B[(size_t)(ka + 1) * Nc + ncl] * nmask;
        // (neg_a, A, neg_b, B, c_mod, C, reuse_a, reuse_b)
        acc = __builtin_amdgcn_wmma_f32_16x16x4_f32(
            false, a, false, b, (short)0, acc, false, false);
    }
    if (K4 < K) {                                    // executes at most once
        const int ka = K4 + (half << 1);
        const bool k0ok = (ka < K);
        const bool k1ok = (ka + 1 < K);
        v2f a, b;
        a.x = k0ok ? Arow[ka]     : 0.0f;
        a.y = k1ok ? Arow[ka + 1] : 0.0f;
        b.x = k0ok ? B[(size_t)ka * Nc + ncl] * nmask       : 0.0f;
        b.y = k1ok ? B[(size_t)(ka + 1) * Nc + ncl] * nmask : 0.0f;
        acc = __builtin_amdgcn_wmma_f32_16x16x4_f32(
            false, a, false, b, (short)0, acc, false, false);
    }

    const float bb = bias ? bias[ncl] : 0.0f;
#pragma unroll
    for (int r = 0; r < 8; ++r) {
        const int mm = row0 + r + (half << 3);
        if (nok && mm < M) {
            float v = acc[r] + bb;
            if (relu) v = fmaxf(v, 0.0f);
            C[(size_t)mm * Nc + n] = v;
        }
    }
}

// ---------------------------------------------------------------------------
// edge scatter (float4):  out[dst] += h[src] * dinv[src] * dinv[dst]
// dq = d/4 feature quads; 1 thread = 1 (edge, quad)
// ---------------------------------------------------------------------------
__global__ void k_agg_edges4(const int* __restrict__ src, const int* __restrict__ dst,
                             const float* __restrict__ dinv,
                             const float4* __restrict__ h,
                             float* __restrict__ out, int E, int dq) {
    long long idx = (long long)blockIdx.x * blockDim.x + threadIdx.x;
    if (idx >= (long long)E * dq) return;
    int e = (int)(idx / dq);
    int f = (int)(idx - (long long)e * dq);
    int s = src[e], t = dst[e];
    float w = dinv[s] * dinv[t];
    float4 v = h[(size_t)s * dq + f];
    float* o = out + ((size_t)t * dq + f) * 4;
    atomicAdd(o + 0, v.x * w);
    atomicAdd(o + 1, v.y * w);
    atomicAdd(o + 2, v.z * w);
    atomicAdd(o + 3, v.w * w);
}

// scalar fallback (d not divisible by 4)
__global__ void k_agg_edges(const int* __restrict__ src, const int* __restrict__ dst,
                            const float* __restrict__ dinv, const float* __restrict__ h,
                            float* __restrict__ out, int E, int d) {
    long long idx = (long long)blockIdx.x * blockDim.x + threadIdx.x;
    if (idx >= (long long)E * d) return;
    int e = (int)(idx / d);
    int f = (int)(idx - (long long)e * d);
    int s = src[e], t = dst[e];
    float w = dinv[s] * dinv[t];
    atomicAdd(&out[(size_t)t * d + f], h[(size_t)s * d + f] * w);
}

// ---------------------------------------------------------------------------
// self loop + bias (+ optional relu):  out = act(out + h*dinv^2 + bias)  (float4)
// ---------------------------------------------------------------------------
__global__ void k_finalize4(const float4* __restrict__ h, const float* __restrict__ dinv,
                            const float4* __restrict__ bias, float4* __restrict__ out,
                            int N, int dq, int relu) {
    long long idx = (long long)blockIdx.x * blockDim.x + threadIdx.x;
    if (idx >= (long long)N * dq) return;
    int i = (int)(idx / dq);
    int f = (int)(idx - (long long)i * dq);
    float w = dinv[i] * dinv[i];
    float4 hv = h[idx];
    float4 ov = out[idx];
    float4 bv = bias[f];
    float4 r;
    r.x = ov.x + hv.x * w + bv.x;
    r.y = ov.y + hv.y * w + bv.y;
    r.z = ov.z + hv.z * w + bv.z;
    r.w = ov.w + hv.w * w + bv.w;
    if (relu) {
        r.x = fmaxf(r.x, 0.0f); r.y = fmaxf(r.y, 0.0f);
        r.z = fmaxf(r.z, 0.0f); r.w = fmaxf(r.w, 0.0f);
    }
    out[idx] = r;
}

__global__ void k_finalize(const float* __restrict__ h, const float* __restrict__ dinv,
                           const float* __restrict__ bias, float* __restrict__ out,
                           int N, int d, int relu) {
    long long idx = (long long)blockIdx.x * blockDim.x + threadIdx.x;
    if (idx >= (long long)N * d) return;
    int i = (int)(idx / d);
    int f = (int)(idx - (long long)i * d);
    float w = dinv[i] * dinv[i];
    float v = out[idx] + h[idx] * w + bias[f];
    if (relu) v = fmaxf(v, 0.0f);
    out[idx] = v;
}

// ---------------------------------------------------------------------------
// row-wise log_softmax over d classes
// ---------------------------------------------------------------------------
__global__ void k_log_softmax(const float* __restrict__ h, float* __restrict__ out,
                              int N, int d) {
    int i = blockIdx.x * blockDim.x + threadIdx.x;
    if (i >= N) return;
    const float* row = h + (size_t)i * d;
    float mx = row[0];
    for (int f = 1; f < d; ++f) mx = fmaxf(mx, row[f]);
    float s = 0.0f;
    for (int f = 0; f < d; ++f) s += expf(row[f] - mx);
    float lse = mx + logf(s);
    for (int f = 0; f < d; ++f) out[(size_t)i * d + f] = row[f] - lse;
}

// ---------------------------------------------------------------------------
// driver
// ---------------------------------------------------------------------------
static inline unsigned blocks_for(long long total, int bs) {
    return (unsigned)((total + bs - 1) / bs);
}

static inline void zero_buf(float* p, long long n, hipStream_t stream) {
    if ((n & 3) == 0) {
        k_fill4<<<blocks_for(n >> 2, 256), 256, 0, stream>>>((float4*)p, 0.0f, (int)(n >> 2));
    } else {
        k_fill<<<blocks_for(n, 256), 256, 0, stream>>>(p, 0.0f, (int)n);
    }
}

static inline void agg(const int* src, const int* dst, const float* dinv,
                       const float* h, float* out, int E, int d, hipStream_t stream) {
    if ((d & 3) == 0) {
        int dq = d >> 2;
        k_agg_edges4<<<blocks_for((long long)E * dq, 256), 256, 0, stream>>>(
            src, dst, dinv, (const float4*)h, out, E, dq);
    } else {
        k_agg_edges<<<blocks_for((long long)E * d, 256), 256, 0, stream>>>(
            src, dst, dinv, h, out, E, d);
    }
}

static inline void finalize(const float* h, const float* dinv, const float* bias,
                            float* out, int N, int d, int relu, hipStream_t stream) {
    if ((d & 3) == 0) {
        int dq = d >> 2;
        k_finalize4<<<blocks_for((long long)N * dq, 256), 256, 0, stream>>>(
            (const float4*)h, dinv, (const float4*)bias, (float4*)out, N, dq, relu);
    } else {
        k_finalize<<<blocks_for((long long)N * d, 256), 256, 0, stream>>>(
            h, dinv, bias, out, N, d, relu);
    }
}

extern "C" void kernel_launch(void* const* d_in, const int* in_sizes, int n_in,
                              void* d_out, int out_size, void* d_ws, size_t ws_size,
                              hipStream_t stream) {
    // setup_inputs order: x, lin_W, lin_b, W1, b1, W2, b2, edge_index
    const float* x     = (const float*)d_in[0];
    const float* lin_W = (const float*)d_in[1];
    const float* lin_b = (const float*)d_in[2];
    const float* W1    = (const float*)d_in[3];
    const float* b1    = (const float*)d_in[4];
    const float* W2    = (const float*)d_in[5];
    const float* b2    = (const float*)d_in[6];
    const int*   eidx  = (const int*)d_in[7];

    const int H1   = in_sizes[2];             // 300
    const int H2   = in_sizes[4];             // 100
    const int Cc   = in_sizes[6];             // 16
    const int F_IN = in_sizes[1] / H1;        // 500
    const int N    = in_sizes[0] / F_IN;      // 50000
    const int E    = in_sizes[7] / 2;         // 800000

    const int* src = eidx;
    const int* dst = eidx + E;

    // workspace layout (floats):
    //   dinv: N
    //   bufA: N*H1   h1      (later reused: h3pre  N*Cc)
    //   bufB: N*H2   h2pre   (later reused: h3agg  N*Cc)
    //   bufC: N*H2   h2 (aggregated + relu)
    float* wsf   = (float*)d_ws;
    float* dinv  = wsf;
    float* bufA  = dinv + N;
    float* bufB  = bufA + (size_t)N * H1;
    float* bufC  = bufB + (size_t)N * H2;
    float* h1    = bufA;
    float* h2pre = bufB;
    float* h2    = bufC;
    float* h3pre = bufA;   // h1 dead after GEMM2
    float* h3agg = bufB;   // h2pre dead after finalize1
    float* outp  = (float*)d_out;

    const int BS = 256;

    // --- degrees & normalization (deg includes self loop) ---
    k_fill<<<blocks_for(N, BS), BS, 0, stream>>>(dinv, 1.0f, N);
    k_count_deg<<<blocks_for(E, BS), BS, 0, stream>>>(dst, dinv, E);
    k_rsqrt_inplace<<<blocks_for(N, BS), BS, 0, stream>>>(dinv, N);

    // --- h1 = relu(x @ lin_W + lin_b) ---
    {
        long long waves = (long long)((N + 15) / 16) * ((H1 + 15) / 16);
        k_gemm_wmma_f32<<<blocks_for(waves * 32, BS), BS, 0, stream>>>(
            x, lin_W, lin_b, h1, N, F_IN, H1, 1);
    }

    // --- GCNConv 1: project, aggregate, self-loop + bias + relu ---
    {
        long long waves = (long long)((N + 15) / 16) * ((H2 + 15) / 16);
        k_gemm_wmma_f32<<<blocks_for(waves * 32, BS), BS, 0, stream>>>(
            h1, W1, nullptr, h2pre, N, H1, H2, 0);
    }
    zero_buf(h2, (long long)N * H2, stream);
    agg(src, dst, dinv, h2pre, h2, E, H2, stream);
    finalize(h2pre, dinv, b1, h2, N, H2, 1, stream);

    // --- GCNConv 2: project, aggregate, self-loop + bias ---
    {
        long long waves = (long long)((N + 15) / 16) * ((Cc + 15) / 16);
        k_gemm_wmma_f32<<<blocks_for(waves * 32, BS), BS, 0, stream>>>(
            h2, W2, nullptr, h3pre, N, H2, Cc, 0);
    }
    zero_buf(h3agg, (long long)N * Cc, stream);
    agg(src, dst, dinv, h3pre, h3agg, E, Cc, stream);
    finalize(h3pre, dinv, b2, h3agg, N, Cc, 0, stream);

    // --- log_softmax ---
    k_log_softmax<<<blocks_for(N, BS), BS, 0, stream>>>(h3agg, outp, N, Cc);
}